// TemporalGNN_44452911514152
// MI455X (gfx1250) — compile-verified
//
#include <hip/hip_runtime.h>
#include <hip/hip_bf16.h>
#include <cstdint>

// ---------------------------------------------------------------------------
// TemporalGNN (EvolveGCNO single step) for MI455X / gfx1250, wave32.
//   W  = GRU(W0, W0)                      (tiny, VALU kernel)
//   xw = x @ W                            (v_wmma_f32_16x16x4_f32)
//   h  = scatter-add of norm * xw[row]    (global_atomic_add_f32, L2 resident)
//   out= relu(h + dinv^2*xw) @ lin_w + b  (self loops folded in here)
// ---------------------------------------------------------------------------

#define FDIM 128

typedef float v2f __attribute__((ext_vector_type(2)));
typedef float v8f __attribute__((ext_vector_type(8)));

__global__ __launch_bounds__(256) void zero_kernel(float* __restrict__ p, long long n) {
  long long i = (long long)blockIdx.x * blockDim.x + threadIdx.x;
  long long st = (long long)gridDim.x * blockDim.x;
  for (; i < n; i += st) p[i] = 0.0f;
}

// One thread per W element: six length-128 dot products + GRU gate math.
__global__ __launch_bounds__(256) void evolve_kernel(
    const float* __restrict__ W0, const float* __restrict__ wih,
    const float* __restrict__ whh, const float* __restrict__ bih,
    const float* __restrict__ bhh, float* __restrict__ W) {
  int t = blockIdx.x * blockDim.x + threadIdx.x;      // 0 .. 128*128-1
  int i = t >> 7;
  int j = t & 127;
  const float4* a4   = (const float4*)(W0  + (size_t)i * FDIM);
  const float4* wri  = (const float4*)(wih + (size_t)(j          ) * FDIM);
  const float4* wzi  = (const float4*)(wih + (size_t)(j +   FDIM ) * FDIM);
  const float4* wni  = (const float4*)(wih + (size_t)(j + 2*FDIM ) * FDIM);
  const float4* wrh  = (const float4*)(whh + (size_t)(j          ) * FDIM);
  const float4* wzh  = (const float4*)(whh + (size_t)(j +   FDIM ) * FDIM);
  const float4* wnh  = (const float4*)(whh + (size_t)(j + 2*FDIM ) * FDIM);
  float ir = 0.f, iz = 0.f, in_ = 0.f, hr = 0.f, hz = 0.f, hn = 0.f;
#pragma unroll 4
  for (int k = 0; k < FDIM / 4; ++k) {
    float4 a = a4[k];
    float4 v;
    v = wri[k]; ir  += a.x*v.x + a.y*v.y + a.z*v.z + a.w*v.w;
    v = wzi[k]; iz  += a.x*v.x + a.y*v.y + a.z*v.z + a.w*v.w;
    v = wni[k]; in_ += a.x*v.x + a.y*v.y + a.z*v.z + a.w*v.w;
    v = wrh[k]; hr  += a.x*v.x + a.y*v.y + a.z*v.z + a.w*v.w;
    v = wzh[k]; hz  += a.x*v.x + a.y*v.y + a.z*v.z + a.w*v.w;
    v = wnh[k]; hn  += a.x*v.x + a.y*v.y + a.z*v.z + a.w*v.w;
  }
  ir  += bih[j];          hr += bhh[j];
  iz  += bih[j + FDIM];   hz += bhh[j + FDIM];
  in_ += bih[j + 2*FDIM]; hn += bhh[j + 2*FDIM];
  float r = 1.0f / (1.0f + expf(-(ir + hr)));
  float z = 1.0f / (1.0f + expf(-(iz + hz)));
  float n = tanhf(in_ + r * hn);
  float w0 = W0[(size_t)i * FDIM + j];
  W[(size_t)i * FDIM + j] = (1.0f - z) * n + z * w0;
}

// deg[col[e]] += w[e]   (self-loop "+1" is added in dinv_kernel)
__global__ __launch_bounds__(256) void degree_kernel(
    const int* __restrict__ col, const float* __restrict__ w,
    float* __restrict__ deg, int E) {
  int i = blockIdx.x * blockDim.x + threadIdx.x;
  int st = gridDim.x * blockDim.x;
  for (; i < E; i += st) unsafeAtomicAdd(&deg[col[i]], w[i]);
}

__global__ __launch_bounds__(256) void dinv_kernel(
    const float* __restrict__ deg, float* __restrict__ dinv, int N) {
  int i = blockIdx.x * blockDim.x + threadIdx.x;
  if (i < N) {
    float d = deg[i] + 1.0f;   // + self loop weight; always > 0
    dinv[i] = rsqrtf(d);
  }
}

// xw = x @ W using V_WMMA_F32_16X16X4_F32.
// Block = 256 threads = 8 waves; each wave owns a 16x16 output tile
// (16 rows of x shared by the block, 16 columns per wave), K looped by 4.
__global__ __launch_bounds__(256) void gemm_kernel(
    const float* __restrict__ x, const float* __restrict__ W,
    float* __restrict__ xw, int N) {
  int lane  = threadIdx.x & 31;
  int wave  = threadIdx.x >> 5;         // 0..7 -> column tile
  int row0  = blockIdx.x * 16;
  int col0  = wave * 16;
  int l16   = lane & 15;
  int khalf = (lane >> 4) * 2;          // 0 for lanes 0-15, 2 for lanes 16-31

  int arow = row0 + l16;
  if (arow >= N) arow = N - 1;          // clamp (keeps EXEC all-ones for WMMA)

  v8f acc = {};
#pragma unroll
  for (int k = 0; k < FDIM; k += 4) {
    // A fragment: 16x4, lane(M=l16) holds K = k+khalf, k+khalf+1
    const float* ap = x + (size_t)arow * FDIM + (k + khalf);
    v2f a;
    a.x = ap[0];
    a.y = ap[1];
    // B fragment: 4x16, lane(N=l16) holds K = k+khalf, k+khalf+1
    const float* bp = W + (size_t)(k + khalf) * FDIM + col0 + l16;
    v2f b;
    b.x = bp[0];
    b.y = bp[FDIM];
    acc = __builtin_amdgcn_wmma_f32_16x16x4_f32(
        /*neg_a=*/false, a, /*neg_b=*/false, b,
        /*c_mod=*/(short)0, acc, /*reuse_a=*/false, /*reuse_b=*/false);
  }
  // D layout: VGPR v -> M = v (+8 for lanes 16-31), N = lane%16
  int mbase = (lane >> 4) * 8;
#pragma unroll
  for (int v = 0; v < 8; ++v) {
    int r = row0 + mbase + v;
    if (r < N) xw[(size_t)r * FDIM + col0 + l16] = acc[v];
  }
}

// One wave per edge: h[col] += norm * xw[row]; each lane owns 4 floats.
__global__ __launch_bounds__(256) void scatter_kernel(
    const int* __restrict__ row, const int* __restrict__ col,
    const float* __restrict__ w, const float* __restrict__ dinv,
    const float* __restrict__ xw, float* __restrict__ h, int E) {
  int gtid = blockIdx.x * blockDim.x + threadIdx.x;
  int lane = gtid & 31;
  int wid  = gtid >> 5;
  int nw   = (gridDim.x * blockDim.x) >> 5;
  for (int e = wid; e < E; e += nw) {
    int r = row[e];
    int c = col[e];
    float nrm = dinv[r] * w[e] * dinv[c];
    const float4* xv = (const float4*)(xw + (size_t)r * FDIM);
    float4 v = xv[lane];
    float* hp = h + (size_t)c * FDIM + lane * 4;
    unsafeAtomicAdd(hp + 0, nrm * v.x);
    unsafeAtomicAdd(hp + 1, nrm * v.y);
    unsafeAtomicAdd(hp + 2, nrm * v.z);
    unsafeAtomicAdd(hp + 3, nrm * v.w);
  }
}

// One wave per node: out[i] = lin_w . relu(h[i] + dinv[i]^2 * xw[i]) + lin_b
__global__ __launch_bounds__(256) void finalize_kernel(
    const float* __restrict__ h, const float* __restrict__ xw,
    const float* __restrict__ dinv, const float* __restrict__ lw,
    const float* __restrict__ lb, float* __restrict__ out, int N) {
  int gtid = blockIdx.x * blockDim.x + threadIdx.x;
  int lane = gtid & 31;
  int wid  = gtid >> 5;
  int nw   = (gridDim.x * blockDim.x) >> 5;
  float4 l4 = ((const float4*)lw)[lane];
  for (int i = wid; i < N; i += nw) {
    float di = dinv[i];
    float d2 = di * di;
    float4 hv = ((const float4*)(h  + (size_t)i * FDIM))[lane];
    float4 xv = ((const float4*)(xw + (size_t)i * FDIM))[lane];
    float s = l4.x * fmaxf(hv.x + d2 * xv.x, 0.0f)
            + l4.y * fmaxf(hv.y + d2 * xv.y, 0.0f)
            + l4.z * fmaxf(hv.z + d2 * xv.z, 0.0f)
            + l4.w * fmaxf(hv.w + d2 * xv.w, 0.0f);
#pragma unroll
    for (int off = 16; off > 0; off >>= 1) s += __shfl_xor(s, off, 32);
    if (lane == 0) out[i] = s + lb[0];
  }
}

extern "C" void kernel_launch(void* const* d_in, const int* in_sizes, int n_in,
                              void* d_out, int out_size, void* d_ws, size_t ws_size,
                              hipStream_t stream) {
  const float* x   = (const float*)d_in[0];
  const int*   ei  = (const int*)  d_in[1];
  const float* ew  = (const float*)d_in[2];
  const float* W0  = (const float*)d_in[3];
  const float* wih = (const float*)d_in[4];
  const float* whh = (const float*)d_in[5];
  const float* bih = (const float*)d_in[6];
  const float* bhh = (const float*)d_in[7];
  const float* lw  = (const float*)d_in[8];
  const float* lb  = (const float*)d_in[9];
  float* out = (float*)d_out;

  const int N = in_sizes[0] / FDIM;
  const int E = in_sizes[1] / 2;
  const int* row = ei;        // edge_index[0] = sources
  const int* col = ei + E;    // edge_index[1] = targets

  // Workspace layout (256-byte aligned chunks).
  char* ws = (char*)d_ws;
  size_t off = 0;
  auto alloc = [&](size_t bytes) -> float* {
    float* p = (float*)(ws + off);
    off += (bytes + 255) & ~(size_t)255;
    return p;
  };
  float* W    = alloc((size_t)FDIM * FDIM * sizeof(float));
  float* deg  = alloc((size_t)N * sizeof(float));
  float* dinv = alloc((size_t)N * sizeof(float));
  float* xw   = alloc((size_t)N * FDIM * sizeof(float));
  float* h    = alloc((size_t)N * FDIM * sizeof(float));
  (void)ws_size;

  // 1) zero accumulators (harness does not re-poison between replays)
  zero_kernel<<<4096, 256, 0, stream>>>(h, (long long)N * FDIM);
  zero_kernel<<<512, 256, 0, stream>>>(deg, (long long)N);

  // 2) W = GRU(W0, W0)
  evolve_kernel<<<(FDIM * FDIM) / 256, 256, 0, stream>>>(W0, wih, whh, bih, bhh, W);

  // 3) degree + dinv
  degree_kernel<<<2048, 256, 0, stream>>>(col, ew, deg, E);
  dinv_kernel<<<(N + 255) / 256, 256, 0, stream>>>(deg, dinv, N);

  // 4) xw = x @ W  (WMMA f32)
  gemm_kernel<<<(N + 15) / 16, 256, 0, stream>>>(x, W, xw, N);

  // 5) edge scatter (L2-resident atomics)
  scatter_kernel<<<4096, 256, 0, stream>>>(row, col, ew, dinv, xw, h, E);

  // 6) self loops + relu + linear
  finalize_kernel<<<4096, 256, 0, stream>>>(h, xw, dinv, lw, lb, out, N);
}